// NegativeBinomial_4741643894906
// MI455X (gfx1250) — compile-verified
//
#include <hip/hip_runtime.h>
#include <math.h>

typedef float v2f __attribute__((ext_vector_type(2)));
typedef float v8f __attribute__((ext_vector_type(8)));

// ---------------------------------------------------------------------------
// Kernel 1: moderator-side reductions + per-group scalar constants.
// consts layout (d_ws): per group g: [r, a=v*sum_m, b=sum_m2, log(a), log(b), lgamma(r)]
// ---------------------------------------------------------------------------
__global__ void nb_prep(const float* __restrict__ mod0,
                        const float* __restrict__ mod1,
                        const float* __restrict__ wm,
                        const float* __restrict__ od_sqrt,
                        int S, int M,
                        float* __restrict__ consts) {
    __shared__ float wmL[256];
    __shared__ float red[256];
    int tid = threadIdx.x;
    for (int i = tid; i < M; i += blockDim.x) wmL[i] = wm[i];
    __syncthreads();

    float sums[4] = {0.f, 0.f, 0.f, 0.f};   // {sum_m0, sum_m2_0, sum_m1, sum_m2_1}
    for (int s = tid; s < S; s += blockDim.x) {
        float d0 = 0.f, d1 = 0.f;
        for (int j = 0; j < M; ++j) {
            float w = wmL[j];
            d0 += mod0[s * M + j] * w;
            d1 += mod1[s * M + j] * w;
        }
        float m0 = expf(d0), m1 = expf(d1);
        sums[0] += m0; sums[1] += m0 * m0;
        sums[2] += m1; sums[3] += m1 * m1;
    }
    for (int q = 0; q < 4; ++q) {
        red[tid] = sums[q];
        __syncthreads();
        for (int st = blockDim.x / 2; st > 0; st >>= 1) {
            if (tid < st) red[tid] += red[tid + st];
            __syncthreads();
        }
        if (tid == 0) sums[q] = red[0];
        __syncthreads();
    }
    if (tid == 0) {
        for (int g = 0; g < 2; ++g) {
            float sm = sums[g * 2], sm2 = sums[g * 2 + 1];
            float sg = od_sqrt[g];
            float v  = 1.f / (sg * sg);
            float r  = v * sm * sm / sm2;   // constant per group
            float a  = v * sm;
            float b  = sm2;
            consts[g * 6 + 0] = r;
            consts[g * 6 + 1] = a;
            consts[g * 6 + 2] = b;
            consts[g * 6 + 3] = logf(a);
            consts[g * 6 + 4] = logf(b);
            consts[g * 6 + 5] = lgammaf(r);
        }
    }
}

// ---------------------------------------------------------------------------
// Two-tile software-pipelined WMMA mat-vec (D known at compile time).
// Per 32-column group: batch 16 global b64 loads (8 per tile, one clause's
// worth in flight) + shared LDS A-fragments, then 16 WMMAs on two
// independent accumulator chains (acc0/acc1) -> deeper MLP and WMMA ILP.
// ---------------------------------------------------------------------------
template <int DD>
__device__ __forceinline__ void pair_matvec(const float* __restrict__ xrow0,
                                            const float* __restrict__ xrow1,
                                            const float* __restrict__ wrow,
                                            int act, v8f& acc0, v8f& acc1) {
    constexpr int U  = 8;               // k-steps per group (32 columns)
    constexpr int NG = DD / (4 * U);
    static_assert(DD % (4 * U) == 0, "D must be a multiple of 32");

    acc0 = v8f{0.f, 0.f, 0.f, 0.f, 0.f, 0.f, 0.f, 0.f};
    acc1 = v8f{0.f, 0.f, 0.f, 0.f, 0.f, 0.f, 0.f, 0.f};

    #pragma unroll
    for (int gp = 0; gp < NG; ++gp) {
        const int kb = gp * 4 * U;
        v2f b0[U], b1[U], a[U];
        #pragma unroll
        for (int u = 0; u < U; ++u) b0[u] = *(const v2f*)(xrow0 + kb + 4 * u);
        #pragma unroll
        for (int u = 0; u < U; ++u) b1[u] = *(const v2f*)(xrow1 + kb + 4 * u);
        #pragma unroll
        for (int u = 0; u < U; ++u) {
            v2f wv = *(const v2f*)(wrow + kb + 4 * u);
            a[u].x = act ? wv.x : 0.f;
            a[u].y = act ? wv.y : 0.f;
        }
        #pragma unroll
        for (int u = 0; u < U; ++u) {
            acc0 = __builtin_amdgcn_wmma_f32_16x16x4_f32(
                       false, a[u], false, b0[u], (short)0, acc0, false, false);
            acc1 = __builtin_amdgcn_wmma_f32_16x16x4_f32(
                       false, a[u], false, b1[u], (short)0, acc1, false, false);
        }
    }
}

// generic fallback (any D % 4 == 0)
__device__ __forceinline__ void pair_matvec_gen(const float* __restrict__ xrow0,
                                                const float* __restrict__ xrow1,
                                                const float* __restrict__ wrow,
                                                int act, int D,
                                                v8f& acc0, v8f& acc1) {
    acc0 = v8f{0.f, 0.f, 0.f, 0.f, 0.f, 0.f, 0.f, 0.f};
    acc1 = v8f{0.f, 0.f, 0.f, 0.f, 0.f, 0.f, 0.f, 0.f};
    for (int kk = 0; kk < D; kk += 4) {
        v2f bf0 = *(const v2f*)(xrow0 + kk);
        v2f bf1 = *(const v2f*)(xrow1 + kk);
        v2f wv  = *(const v2f*)(wrow + kk);
        v2f af;
        af.x = act ? wv.x : 0.f;
        af.y = act ? wv.y : 0.f;
        acc0 = __builtin_amdgcn_wmma_f32_16x16x4_f32(
                   false, af, false, bf0, (short)0, acc0, false, false);
        acc1 = __builtin_amdgcn_wmma_f32_16x16x4_f32(
                   false, af, false, bf1, (short)0, acc1, false, false);
    }
}

// fused negative-binomial per-voxel term
__device__ __forceinline__ float nb_contrib(float t, int vox, int V,
                                            const float* __restrict__ yptr,
                                            float rC, float aC, float bC,
                                            float logA, float logB, float lgR) {
    int voxc = vox < V ? vox : V - 1;
    float y  = yptr[voxc];
    float mu    = expf(t);
    float denom = aC + mu * bC;                 // v*sum_m + mu*sum_m2
    float ldn   = logf(denom);
    float c = rC * (logA - ldn)                 // r * log(1-p)
            + y * (t + logB - ldn);             // y * log(p)
    if (y > 0.f)
        c += lgammaf(y + rC) - lgammaf(y + 1.f) - lgR;
    return (vox < V) ? c : 0.f;
}

// ---------------------------------------------------------------------------
// Kernel 2: the 512 MB streaming pass. One wave owns TWO 16-voxel tiles.
//   A (16x4) = weights: rows {0,8}=w0, rows {1,9}=w1, others 0  (from LDS)
//   B (4x16) = X chunk: B[k][n] = X[base+n][kk+k]                (from HBM)
//   C row0/1 -> t0/t1 on lanes 0-15 ; C row8/9 -> t0/t1 on lanes 16-31.
// Lanes 0-15 run the group-0 epilogue, lanes 16-31 group-1: all 32 lanes busy.
// ---------------------------------------------------------------------------
__global__ void nb_main(const float* __restrict__ X,
                        const float* __restrict__ foci0,
                        const float* __restrict__ foci1,
                        const float* __restrict__ spat_coef,   // [2, D]
                        const float* __restrict__ consts,
                        float* __restrict__ partials,
                        int V, int D) {
    __shared__ float wlds[512];   // D <= 256, 2 weight vectors
    __shared__ float red[256];
    int tid = threadIdx.x;
    for (int i = tid; i < 2 * D; i += blockDim.x) wlds[i] = spat_coef[i];
    __syncthreads();

    const int lane = tid & 31;                 // wave32
    const int n    = lane & 15;                // voxel-in-tile / A row
    const int h    = lane >> 4;                // K-half, doubles as group id
    const int g    = h;

    // per-lane group constants
    const float rC   = consts[g * 6 + 0];
    const float aC   = consts[g * 6 + 1];
    const float bC   = consts[g * 6 + 2];
    const float logA = consts[g * 6 + 3];
    const float logB = consts[g * 6 + 4];
    const float lgR  = consts[g * 6 + 5];
    const float* yptr = g ? foci1 : foci0;

    // A-fragment selection: active rows m in {0,1,8,9}; widx picks w0/w1
    const int act  = ((n & 7) < 2) ? 1 : 0;
    const int widx = n & 1;
    const int koff = 2 * h;
    const float* wrow = &wlds[widx * D + koff];

    const int wavesPerBlock = blockDim.x >> 5;
    const int gwave  = blockIdx.x * wavesPerBlock + (tid >> 5);
    const int nwaves = gridDim.x * wavesPerBlock;
    const int ntiles = (V + 15) >> 4;
    const int npairs = (ntiles + 1) >> 1;

    float local = 0.f;
    for (int pair = gwave; pair < npairs; pair += nwaves) {
        const int base0 = (pair * 2) << 4;
        const int base1 = base0 + 16;
        int row0 = base0 + n;
        int row1 = base1 + n;
        int rc0  = row0 < V ? row0 : V - 1;        // clamp keeps EXEC full
        int rc1  = row1 < V ? row1 : V - 1;
        const float* xrow0 = X + (size_t)rc0 * D + koff;
        const float* xrow1 = X + (size_t)rc1 * D + koff;

        v8f acc0, acc1;
        if (D == 256) pair_matvec<256>(xrow0, xrow1, wrow, act, acc0, acc1);
        else          pair_matvec_gen(xrow0, xrow1, wrow, act, D, acc0, acc1);

        // lanes 0-15: t = C[g? 1:0][n]; lanes 16-31: t = C[g? 9:8][n]
        float t0 = g ? acc0[1] : acc0[0];
        float t1 = g ? acc1[1] : acc1[0];
        local += nb_contrib(t0, base0 + n, V, yptr, rC, aC, bC, logA, logB, lgR);
        local += nb_contrib(t1, base1 + n, V, yptr, rC, aC, bC, logA, logB, lgR);
    }

    red[tid] = local;
    __syncthreads();
    for (int st = blockDim.x / 2; st > 0; st >>= 1) {
        if (tid < st) red[tid] += red[tid + st];
        __syncthreads();
    }
    if (tid == 0) partials[blockIdx.x] = red[0];
}

// ---------------------------------------------------------------------------
// Kernel 3: deterministic final reduction, out = -(ll0 + ll1)
// ---------------------------------------------------------------------------
__global__ void nb_final(const float* __restrict__ partials, int nblocks,
                         float* __restrict__ out) {
    __shared__ float red[256];
    int tid = threadIdx.x;
    float s = 0.f;
    for (int i = tid; i < nblocks; i += blockDim.x) s += partials[i];
    red[tid] = s;
    __syncthreads();
    for (int st = blockDim.x / 2; st > 0; st >>= 1) {
        if (tid < st) red[tid] += red[tid + st];
        __syncthreads();
    }
    if (tid == 0) out[0] = -red[0];
}

extern "C" void kernel_launch(void* const* d_in, const int* in_sizes, int n_in,
                              void* d_out, int out_size, void* d_ws, size_t ws_size,
                              hipStream_t stream) {
    const float* X    = (const float*)d_in[0];   // [V, D]
    const float* mod0 = (const float*)d_in[1];   // [S, M]
    const float* mod1 = (const float*)d_in[2];
    const float* y0   = (const float*)d_in[3];   // [V]
    const float* y1   = (const float*)d_in[4];
    const float* spat = (const float*)d_in[5];   // [2, D]
    const float* wm   = (const float*)d_in[6];   // [M]
    const float* od   = (const float*)d_in[7];   // [2]
    float* out = (float*)d_out;

    const int M = in_sizes[6];
    const int S = in_sizes[1] / M;
    const int D = in_sizes[5] / 2;
    const int V = in_sizes[3];

    float* consts   = (float*)d_ws;              // 12 floats
    float* partials = (float*)d_ws + 16;         // block partials

    const int ntiles = (V + 15) / 16;
    const int npairs = (ntiles + 1) / 2;
    const int threads = 256;                     // 8 wave32s
    int blocks = (npairs + 7) / 8;
    int maxBlocks = (int)(ws_size / sizeof(float)) - 16;
    if (maxBlocks < 1) maxBlocks = 1;
    if (blocks > maxBlocks) blocks = maxBlocks;
    if (blocks > 8192) blocks = 8192;

    nb_prep<<<1, threads, 0, stream>>>(mod0, mod1, wm, od, S, M, consts);
    nb_main<<<blocks, threads, 0, stream>>>(X, y0, y1, spat, consts, partials, V, D);
    nb_final<<<1, threads, 0, stream>>>(partials, blocks, out);
}